// Attention_18133351924379
// MI455X (gfx1250) — compile-verified
//
#include <hip/hip_runtime.h>
#include <math.h>

typedef __attribute__((ext_vector_type(2))) float v2f;
typedef __attribute__((ext_vector_type(4))) float v4f;
typedef __attribute__((ext_vector_type(8))) float v8f;

#define BATCH 1026
#define DIM   256
#define QLEN  512

// One workgroup (256 threads, 8 waves) per batch element.
// Single pass over context: WMMA scores + online-softmax accumulation.
__global__ __launch_bounds__(256) void attn_kernel(
    const float* __restrict__ query,    // [B,1,D]
    const float* __restrict__ context,  // [B,Q,D]
    const float* __restrict__ W_in,     // [D,D]
    const float* __restrict__ W_out,    // [D,2D]
    const float* __restrict__ ae,       // [B]
    const float* __restrict__ ab,       // [B]
    float* __restrict__ out,            // [B,D]
    float* __restrict__ wout)           // [B,Q]
{
    __shared__ float s_query[DIM];
    __shared__ float s_q[DIM];
    __shared__ float s_scores[QLEN];     // raw scores
    __shared__ float s_bt[QLEN];
    __shared__ float s_mix[DIM];
    __shared__ float s_acc1[8 * DIM];    // per-wave scaled S1
    __shared__ float s_acc2[8 * DIM];    // per-wave scaled S2
    __shared__ float s_wm[8];
    __shared__ float s_wz[8];

    const int tid  = threadIdx.x;        // 0..255
    const int b    = blockIdx.x;
    const int lane = tid & 31;
    const int wave = tid >> 5;           // 0..7
    const int half = lane >> 4;          // 0 or 1
    const int mlan = lane & 15;

    const float abv = ab[b];
    const float aev = ae[b];

    // ---- stage query row + bt table ----
    s_query[tid] = query[(size_t)b * DIM + tid];
    s_bt[tid]        = __expf(-abv * (float)(QLEN - 1 - tid));
    s_bt[tid + 256]  = __expf(-abv * (float)(255 - tid));
    __syncthreads();

    // ---- pass 0: q = query . W_in^T (thread e owns element e; W_in L2-resident) ----
    {
        float acc = 0.f;
        const float* wrow = W_in + (size_t)tid * DIM;
        #pragma unroll 8
        for (int k = 0; k < DIM; ++k) acc = fmaf(wrow[k], s_query[k], acc);
        s_q[tid] = acc;
    }
    __syncthreads();

    // ---- fused pass: per wave, 4 tiles of 16 rows; single read of context ----
    float m_run = -INFINITY;
    float Z = 0.f;
    float S1[8], S2[8];
    #pragma unroll
    for (int j = 0; j < 8; ++j) { S1[j] = 0.f; S2[j] = 0.f; }

    for (int t = 0; t < 4; ++t) {
        const int m0 = wave * 64 + t * 16;

        // --- scores for this tile via V_WMMA_F32_16X16X4_F32 ---
        // A fragment (16x4 f32): lane L, vgpr v <-> M=L%16, K=v+2*(L/16)
        const float* arow =
            context + ((size_t)b * QLEN + m0 + mlan) * DIM + (half << 1);
        if (t < 3) __builtin_prefetch(arow + 16 * DIM, 0, 1);  // next tile
        v8f acc = {0.f, 0.f, 0.f, 0.f, 0.f, 0.f, 0.f, 0.f};
        #pragma unroll 8
        for (int k0 = 0; k0 < DIM; k0 += 4) {
            v2f av = *(const v2f*)(arow + k0);
            const int kb = k0 + (half << 1);
            v2f bv = { s_q[kb], s_q[kb + 1] };
            acc = __builtin_amdgcn_wmma_f32_16x16x4_f32(
                false, av, false, bv, (short)0, acc, false, false);
        }
        // D layout: vgpr r, lane 0 -> M=r, lane 16 -> M=8+r; all N columns equal.
        if (mlan == 0) {
            const int base = m0 + half * 8;
            #pragma unroll
            for (int r = 0; r < 8; ++r) s_scores[base + r] = acc[r];
        }
        // same-wave LDS RAW: DS ops are in-order within a wave.

        // --- online softmax update (wave-uniform) ---
        float tmax = -INFINITY;
        #pragma unroll
        for (int r = 0; r < 16; ++r) tmax = fmaxf(tmax, s_scores[m0 + r]);
        const float newm  = fmaxf(m_run, tmax);
        const float scale = __expf(m_run - newm);   // 0 on first tile
        Z *= scale;
        #pragma unroll
        for (int j = 0; j < 8; ++j) { S1[j] *= scale; S2[j] *= scale; }

        // --- accumulate: re-read tile (L2/WGP$ hit; last use -> non-temporal) ---
        const float* crow =
            context + ((size_t)b * QLEN + m0) * DIM + lane * 8;  // lane owns 8 d's
        #pragma unroll 4
        for (int r = 0; r < 16; ++r) {
            const float e  = __expf(s_scores[m0 + r] - newm);
            const float c2 = e * s_bt[m0 + r];
            Z += e;
            v4f ca = __builtin_nontemporal_load((const v4f*)crow);
            v4f cb = __builtin_nontemporal_load((const v4f*)(crow + 4));
            #pragma unroll
            for (int j = 0; j < 4; ++j) {
                S1[j]     = fmaf(e,  ca[j], S1[j]);
                S2[j]     = fmaf(c2, fmaxf(aev * ca[j], 0.f), S2[j]);
                S1[4 + j] = fmaf(e,  cb[j], S1[4 + j]);
                S2[4 + j] = fmaf(c2, fmaxf(aev * cb[j], 0.f), S2[4 + j]);
            }
            crow += DIM;
        }
        m_run = newm;
    }

    // ---- cross-wave combine ----
    if (lane == 0) s_wm[wave] = m_run;
    __syncthreads();
    float M = -INFINITY;
    #pragma unroll
    for (int w = 0; w < 8; ++w) M = fmaxf(M, s_wm[w]);
    const float f = __expf(m_run - M);
    #pragma unroll
    for (int j = 0; j < 8; ++j) {
        s_acc1[wave * DIM + lane * 8 + j] = S1[j] * f;
        s_acc2[wave * DIM + lane * 8 + j] = S2[j] * f;
    }
    if (lane == 0) s_wz[wave] = Z * f;
    __syncthreads();

    float Ztot = 0.f;
    #pragma unroll
    for (int w = 0; w < 8; ++w) Ztot += s_wz[w];
    const float inv_z = 1.0f / Ztot;

    // mix_sum[d] = (S1_tot + S2_tot) / Z
    {
        float a = 0.f;
        #pragma unroll
        for (int w = 0; w < 8; ++w)
            a += s_acc1[w * DIM + tid] + s_acc2[w * DIM + tid];
        s_mix[tid] = a * inv_z;
    }
    // weights output: w_q = exp(s_q - M) / Z
    wout[(size_t)b * QLEN + tid]       = __expf(s_scores[tid] - M) * inv_z;
    wout[(size_t)b * QLEN + tid + 256] = __expf(s_scores[tid + 256] - M) * inv_z;
    __syncthreads();

    // ---- pass 3: out[d] = tanh( [mix_sum; q] . W_out[d,:] )  (W_out L2-resident) ----
    {
        const float* wrow = W_out + (size_t)tid * (2 * DIM);
        float acc = 0.f;
        #pragma unroll 8
        for (int c = 0; c < DIM; ++c) acc = fmaf(s_mix[c], wrow[c], acc);
        #pragma unroll 8
        for (int c = 0; c < DIM; ++c) acc = fmaf(s_q[c], wrow[DIM + c], acc);
        out[(size_t)b * DIM + tid] = tanhf(acc);
    }
}

extern "C" void kernel_launch(void* const* d_in, const int* in_sizes, int n_in,
                              void* d_out, int out_size, void* d_ws, size_t ws_size,
                              hipStream_t stream) {
    const float* query   = (const float*)d_in[0];
    const float* context = (const float*)d_in[1];
    const float* W_in    = (const float*)d_in[2];
    const float* W_out   = (const float*)d_in[3];
    const float* ae      = (const float*)d_in[4];
    const float* ab      = (const float*)d_in[5];

    float* out  = (float*)d_out;                       // [B, D]
    float* wout = (float*)d_out + (size_t)BATCH * DIM; // [B, Q]

    attn_kernel<<<BATCH, 256, 0, stream>>>(query, context, W_in, W_out,
                                           ae, ab, out, wout);
}